// GaussianMixtureModel_35261681500402
// MI455X (gfx1250) — compile-verified
//
#include <hip/hip_runtime.h>
#include <hip/hip_bf16.h>
#include <math.h>

typedef __attribute__((ext_vector_type(2))) float v2f;
typedef __attribute__((ext_vector_type(8))) float v8f;

#define GMM_D 64
#define GMM_K 16

// ---------------------------------------------------------------------------
// Phase 1: per component k (one block each, 64 threads):
//   L = chol(cov_k); W = L^{-1}; v = W*mu_k;
//   cterm_k = log(w_k) - sum(log diag L) - (d/2) log(2*pi)
// ---------------------------------------------------------------------------
__global__ __launch_bounds__(64) void gmm_prepare(
    const float* __restrict__ cov, const float* __restrict__ means,
    const float* __restrict__ weights,
    float* __restrict__ Wout, float* __restrict__ vout, float* __restrict__ cterm)
{
  __shared__ float sA[GMM_D * GMM_D];
  __shared__ float sW[GMM_D * GMM_D];
  const int k = blockIdx.x;
  const int t = threadIdx.x;

  const float* A = cov + (size_t)k * GMM_D * GMM_D;
  for (int i = 0; i < GMM_D; ++i) sA[i * GMM_D + t] = A[i * GMM_D + t];
  __syncthreads();

  // Right-looking Cholesky on lower triangle (thread t owns row t).
  for (int j = 0; j < GMM_D; ++j) {
    if (t == 0) sA[j * GMM_D + j] = sqrtf(sA[j * GMM_D + j]);
    __syncthreads();
    if (t > j) sA[t * GMM_D + j] /= sA[j * GMM_D + j];
    __syncthreads();
    if (t > j) {
      float Lij = sA[t * GMM_D + j];
      for (int c = j + 1; c <= t; ++c) sA[t * GMM_D + c] -= Lij * sA[c * GMM_D + j];
    }
    __syncthreads();
  }

  // W = L^{-1}: thread t computes column t by forward substitution (thread-local
  // dependency chain; LDS ops from the same wave are in-order).
  for (int i = 0; i < GMM_D; ++i) {
    if (i < t) { sW[i * GMM_D + t] = 0.0f; continue; }
    float s = (i == t) ? 1.0f : 0.0f;
    for (int j = t; j < i; ++j) s -= sA[i * GMM_D + j] * sW[j * GMM_D + t];
    sW[i * GMM_D + t] = s / sA[i * GMM_D + i];
  }
  __syncthreads();

  float* Wk = Wout + (size_t)k * GMM_D * GMM_D;
  for (int i = 0; i < GMM_D; ++i) Wk[i * GMM_D + t] = sW[i * GMM_D + t];  // coalesced

  const float* mu = means + k * GMM_D;
  float acc = 0.0f;
  for (int j = 0; j < GMM_D; ++j) acc += sW[t * GMM_D + j] * mu[j];
  vout[k * GMM_D + t] = acc;

  if (t == 0) {
    float hld = 0.0f;
    for (int j = 0; j < GMM_D; ++j) hld += logf(sA[j * GMM_D + j]);
    const float log2pi = 1.8378770664093453f;
    cterm[k] = logf(weights[k]) - hld - 0.5f * (float)GMM_D * log2pi;
  }
}

// ---------------------------------------------------------------------------
// Phase 2: each wave handles 2 slabs of 16 samples. Per component:
//   acc (init -v) += W_k[16x4 tile] x X[4x16 tile]  via v_wmma_f32_16x16x4_f32
//   maha = column sums of acc^2; streaming logsumexp over k.
// Fragment layouts per CDNA5 ISA 7.12.2 (32-bit A 16x4, B 4x16, C/D 16x16):
//   A: lane l (M = l&15) holds K = {2*(l>>4), 2*(l>>4)+1} in v0,v1 -> b64 load
//   B: lane l (N = l&15) holds K = {2*(l>>4), 2*(l>>4)+1} in v0,v1 -> b64 load
//   D: VGPR r holds row M = r + 8*(l>>4), column N = l&15
// ---------------------------------------------------------------------------
__global__ __launch_bounds__(256) void gmm_phase2(
    const float* __restrict__ data, const float* __restrict__ W,
    const float* __restrict__ v, const float* __restrict__ cterm,
    float* __restrict__ partial, int nSamples, int nWaveTasks)
{
  const int wave = (int)((blockIdx.x * blockDim.x + threadIdx.x) >> 5);
  const int lane = (int)(threadIdx.x & 31);
  if (wave >= nWaveTasks) return;  // wave-uniform: EXEC stays all-1s for WMMA
  const int lh  = lane >> 4;
  const int l15 = lane & 15;

  const int n0 = wave * 32;
  int idx0 = n0 + l15;       if (idx0 > nSamples - 1) idx0 = nSamples - 1;
  int idx1 = n0 + 16 + l15;  if (idx1 > nSamples - 1) idx1 = nSamples - 1;
  const float* bbase0 = data + (size_t)idx0 * GMM_D + 2 * lh;
  const float* bbase1 = data + (size_t)idx1 * GMM_D + 2 * lh;

  float mx0 = -__builtin_inff(), mx1 = -__builtin_inff();
  float sm0 = 0.0f, sm1 = 0.0f;

  for (int k = 0; k < GMM_K; ++k) {
    const float* Wk = W + (size_t)k * GMM_D * GMM_D;
    const float* vk = v + k * GMM_D;

    v8f acc0[4], acc1[4];
#pragma unroll
    for (int m = 0; m < 4; ++m) {
      const float* vp = vk + m * 16 + 8 * lh;
      v8f t;
#pragma unroll
      for (int r = 0; r < 8; ++r) t[r] = -vp[r];
      acc0[m] = t;
      acc1[m] = t;
    }

    const float* a0 = Wk + (0 * 16 + l15) * GMM_D + 2 * lh;
    const float* a1 = Wk + (1 * 16 + l15) * GMM_D + 2 * lh;
    const float* a2 = Wk + (2 * 16 + l15) * GMM_D + 2 * lh;
    const float* a3 = Wk + (3 * 16 + l15) * GMM_D + 2 * lh;

#pragma unroll
    for (int kk = 0; kk < GMM_D; kk += 4) {
      v2f b0 = *(const v2f*)(bbase0 + kk);
      v2f b1 = *(const v2f*)(bbase1 + kk);
      v2f af;
      af = *(const v2f*)(a0 + kk);
      acc0[0] = __builtin_amdgcn_wmma_f32_16x16x4_f32(false, af, false, b0, (short)0, acc0[0], false, false);
      acc1[0] = __builtin_amdgcn_wmma_f32_16x16x4_f32(false, af, false, b1, (short)0, acc1[0], false, false);
      af = *(const v2f*)(a1 + kk);
      acc0[1] = __builtin_amdgcn_wmma_f32_16x16x4_f32(false, af, false, b0, (short)0, acc0[1], false, false);
      acc1[1] = __builtin_amdgcn_wmma_f32_16x16x4_f32(false, af, false, b1, (short)0, acc1[1], false, false);
      af = *(const v2f*)(a2 + kk);
      acc0[2] = __builtin_amdgcn_wmma_f32_16x16x4_f32(false, af, false, b0, (short)0, acc0[2], false, false);
      acc1[2] = __builtin_amdgcn_wmma_f32_16x16x4_f32(false, af, false, b1, (short)0, acc1[2], false, false);
      af = *(const v2f*)(a3 + kk);
      acc0[3] = __builtin_amdgcn_wmma_f32_16x16x4_f32(false, af, false, b0, (short)0, acc0[3], false, false);
      acc1[3] = __builtin_amdgcn_wmma_f32_16x16x4_f32(false, af, false, b1, (short)0, acc1[3], false, false);
    }

    float p0 = 0.0f, p1 = 0.0f;
#pragma unroll
    for (int m = 0; m < 4; ++m) {
#pragma unroll
      for (int r = 0; r < 8; ++r) {
        p0 += acc0[m][r] * acc0[m][r];
        p1 += acc1[m][r] * acc1[m][r];
      }
    }
    // Combine the two lane halves (rows 0..7+base vs 8..15+base of each tile).
    p0 += __shfl_xor(p0, 16, 32);
    p1 += __shfl_xor(p1, 16, 32);

    const float c = cterm[k];
    const float lp0 = c - 0.5f * p0;
    const float lp1 = c - 0.5f * p1;

    float nm0 = fmaxf(mx0, lp0);
    sm0 = sm0 * __expf(mx0 - nm0) + __expf(lp0 - nm0);
    mx0 = nm0;
    float nm1 = fmaxf(mx1, lp1);
    sm1 = sm1 * __expf(mx1 - nm1) + __expf(lp1 - nm1);
    mx1 = nm1;
  }

  float lse0 = mx0 + __logf(sm0);
  float lse1 = mx1 + __logf(sm1);
  if (n0 + l15 >= nSamples)      lse0 = 0.0f;
  if (n0 + 16 + l15 >= nSamples) lse1 = 0.0f;

  float tot = lse0 + lse1;
#pragma unroll
  for (int off = 16; off > 0; off >>= 1) tot += __shfl_xor(tot, off, 32);
  tot *= 0.5f;  // each sample's lse is duplicated in both lane halves
  if (lane == 0) partial[wave] = tot;
}

// ---------------------------------------------------------------------------
// Phase 3: deterministic fixed-order reduction of wave partials (double acc).
// ---------------------------------------------------------------------------
__global__ __launch_bounds__(256) void gmm_reduce(
    const float* __restrict__ partial, int n, float* __restrict__ out)
{
  __shared__ double sbuf[256];
  double s = 0.0;
  for (int i = threadIdx.x; i < n; i += 256) s += (double)partial[i];
  sbuf[threadIdx.x] = s;
  __syncthreads();
  for (int off = 128; off > 0; off >>= 1) {
    if ((int)threadIdx.x < off) sbuf[threadIdx.x] += sbuf[threadIdx.x + off];
    __syncthreads();
  }
  if (threadIdx.x == 0) out[0] = (float)sbuf[0];
}

// ---------------------------------------------------------------------------
extern "C" void kernel_launch(void* const* d_in, const int* in_sizes, int n_in,
                              void* d_out, int out_size, void* d_ws, size_t ws_size,
                              hipStream_t stream) {
  const float* data    = (const float*)d_in[0];  // [N, 64]
  const float* weights = (const float*)d_in[1];  // [16]
  const float* means   = (const float*)d_in[2];  // [16, 64]
  const float* cov     = (const float*)d_in[3];  // [16, 64, 64]
  const int N = in_sizes[0] / GMM_D;

  float* wsf     = (float*)d_ws;
  float* Wp      = wsf;                                   // 16*64*64 floats
  float* vp      = Wp + GMM_K * GMM_D * GMM_D;            // 16*64 floats
  float* ct      = vp + GMM_K * GMM_D;                    // 16 floats
  float* partial = ct + GMM_K;                            // nWaves floats

  const int nWaves = (N + 31) / 32;
  const int blocks = (nWaves + 7) / 8;  // 8 waves (256 threads) per block

  gmm_prepare<<<GMM_K, GMM_D, 0, stream>>>(cov, means, weights, Wp, vp, ct);
  gmm_phase2<<<blocks, 256, 0, stream>>>(data, Wp, vp, ct, partial, N, nWaves);
  gmm_reduce<<<1, 256, 0, stream>>>(partial, nWaves, (float*)d_out);
}